// RPN_64879775973768
// MI455X (gfx1250) — compile-verified
//
#include <hip/hip_runtime.h>
#include <hip/hip_bf16.h>

typedef __attribute__((ext_vector_type(2))) float v2f;
typedef __attribute__((ext_vector_type(8))) float v8f;

#define KTOT 2304          // 256*9 reduction length of 3x3 conv
#define KBLK 36            // K-slab staged in LDS (9 WMMA K=4 steps)
#define NANC 36864         // 64*64*9 anchors
#define NG   16            // ground-truth boxes

// anchor (h, w) per anchor-type a = si*3+ri, scales {4,8,16}, ratios {0.5,1,2}
__device__ const float AH[9] = {
    5.656854249492381f, 4.f, 2.8284271247461903f,
    11.313708498984761f, 8.f, 5.656854249492381f,
    22.627416997969522f, 16.f, 11.313708498984761f};
__device__ const float AW[9] = {
    2.8284271247461903f, 4.f, 5.656854249492381f,
    5.656854249492381f, 8.f, 11.313708498984761f,
    11.313708498984761f, 16.f, 22.627416997969522f};

// ---------------------------------------------------------------------------
// Implicit-GEMM 3x3 conv via V_WMMA_F32_16X16X4_F32.
// Block = 256 threads = 8 waves, covers 128 couts x 64 pixels (one image row).
// Wave = 16 couts x 64 pixels (4 accumulator tiles) so each A (weight)
// fragment load feeds 4 WMMAs -> 4x less L2 weight traffic.
// mode 0: all 16 batches, reduce accumulators to per-channel (sum,sumsq) partials
// mode 1: batch 0 only, fused BN (precomputed scale/shift) + ReLU -> h0[256][4096]
// ---------------------------------------------------------------------------
__global__ __launch_bounds__(256) void conv_wmma_kernel(
    const float* __restrict__ x, const float* __restrict__ cw,
    const float* __restrict__ cb,
    const float* __restrict__ scaleA, const float* __restrict__ shiftA,
    float* __restrict__ sumP, float* __restrict__ sqP,
    float* __restrict__ h0, int mode)
{
  __shared__ float smem[8192];   // K-slab (36x64) / reduce buffer (8x1024)
  const int tid  = threadIdx.x;
  const int lane = tid & 31;
  const int wv   = tid >> 5;
  const int bi   = blockIdx.x;
  int halfI, prow, b;
  if (mode == 0) { halfI = bi & 1; prow = (bi >> 1) & 63; b = bi >> 7; }
  else           { halfI = bi & 1; prow = bi >> 1;        b = 0;       }

  const int cout0 = halfI * 128 + wv * 16;
  const int m     = lane & 15;     // M row (A) / N col (B)
  const int gsel  = lane >> 4;     // lane half -> K pair select
  const float* wrow = cw + (size_t)(cout0 + m) * KTOT;
  const float* xb = x + (size_t)b * 256 * 4096;

  v8f c0 = {}, c1 = {}, c2 = {}, c3 = {};
  for (int kb = 0; kb < KTOT; kb += KBLK) {
    // stage im2col slab B[36][64] into LDS (zero-padded borders)
    for (int e = tid; e < KBLK * 64; e += 256) {
      int krow = e >> 6;
      int pix  = e & 63;
      int k    = kb + krow;
      int cin  = k / 9;
      int r    = k - cin * 9;
      int ky   = r / 3;
      int kx   = r - ky * 3;
      int iy   = prow + ky - 1;
      int ix   = pix + kx - 1;
      float val = 0.f;
      if ((unsigned)iy < 64u && (unsigned)ix < 64u)
        val = xb[((size_t)cin << 12) + (iy << 6) + ix];
      smem[e] = val;
    }
    __syncthreads();
#pragma unroll
    for (int s = 0; s < KBLK / 4; ++s) {
      // A 16x4 f32: v0 = K {0,2}, v1 = K {1,3} split across lane halves
      v2f a = *reinterpret_cast<const v2f*>(wrow + kb + s * 4 + gsel * 2);
      int brow = s * 4 + gsel * 2;
      const float* bp = &smem[brow * 64 + m];
      v2f b0; b0.x = bp[0];  b0.y = bp[64];
      c0 = __builtin_amdgcn_wmma_f32_16x16x4_f32(false, a, false, b0, (short)0, c0, false, false);
      v2f b1; b1.x = bp[16]; b1.y = bp[80];
      c1 = __builtin_amdgcn_wmma_f32_16x16x4_f32(false, a, false, b1, (short)0, c1, false, false);
      v2f b2; b2.x = bp[32]; b2.y = bp[96];
      c2 = __builtin_amdgcn_wmma_f32_16x16x4_f32(false, a, false, b2, (short)0, c2, false, false);
      v2f b3; b3.x = bp[48]; b3.y = bp[112];
      c3 = __builtin_amdgcn_wmma_f32_16x16x4_f32(false, a, false, b3, (short)0, c3, false, false);
    }
    __syncthreads();
  }

  // bias (conv output y = conv + b[cout])
#pragma unroll
  for (int v = 0; v < 8; ++v) {
    float bc = cb[cout0 + v + 8 * gsel];
    c0[v] += bc; c1[v] += bc; c2[v] += bc; c3[v] += bc;
  }

  if (mode == 1) {
    // fused batchnorm (scale/shift precomputed) + ReLU -> h0
    const int p0 = prow * 64;
#pragma unroll
    for (int v = 0; v < 8; ++v) {
      int co = cout0 + v + 8 * gsel;
      float sc = scaleA[co], sf = shiftA[co];
      float* hp = h0 + ((size_t)co << 12) + p0 + m;
      hp[0]  = fmaxf(0.f, c0[v] * sc + sf);
      hp[16] = fmaxf(0.f, c1[v] * sc + sf);
      hp[32] = fmaxf(0.f, c2[v] * sc + sf);
      hp[48] = fmaxf(0.f, c3[v] * sc + sf);
    }
  } else {
    // reduce tiles to per-channel (sum, sumsq) over their 64 pixels
#pragma unroll
    for (int v = 0; v < 8; ++v) {
      smem[wv * 1024 +       v * 32 + lane] = c0[v];
      smem[wv * 1024 + 256 + v * 32 + lane] = c1[v];
      smem[wv * 1024 + 512 + v * 32 + lane] = c2[v];
      smem[wv * 1024 + 768 + v * 32 + lane] = c3[v];
    }
    __syncthreads();
    if (tid < 128) {
      int w8 = tid >> 4, mloc = tid & 15;
      int co = halfI * 128 + w8 * 16 + mloc;
      int v  = mloc & 7, g2 = mloc >> 3;
      float s = 0.f, q = 0.f;
#pragma unroll
      for (int st = 0; st < 4; ++st) {
        const float* src = &smem[w8 * 1024 + st * 256 + v * 32 + g2 * 16];
#pragma unroll
        for (int i = 0; i < 16; ++i) { float t = src[i]; s += t; q += t * t; }
      }
      int slot = b * 64 + prow;
      sumP[((size_t)co << 10) + slot] = s;
      sqP [((size_t)co << 10) + slot] = q;
    }
  }
}

// Reduce 1024 partials per channel -> BN scale/shift
__global__ __launch_bounds__(256) void stats_kernel(
    const float* __restrict__ sumP, const float* __restrict__ sqP,
    const float* __restrict__ gamma, const float* __restrict__ beta,
    float* __restrict__ scaleA, float* __restrict__ shiftA)
{
  __shared__ float ss[256], qq[256];
  int cI = blockIdx.x;
  float s = 0.f, q = 0.f;
  for (int i = threadIdx.x; i < 1024; i += 256) {
    s += sumP[((size_t)cI << 10) + i];
    q += sqP [((size_t)cI << 10) + i];
  }
  ss[threadIdx.x] = s; qq[threadIdx.x] = q;
  __syncthreads();
  for (int off = 128; off > 0; off >>= 1) {
    if (threadIdx.x < off) {
      ss[threadIdx.x] += ss[threadIdx.x + off];
      qq[threadIdx.x] += qq[threadIdx.x + off];
    }
    __syncthreads();
  }
  if (threadIdx.x == 0) {
    float mean = ss[0] * (1.f / 65536.f);
    float var  = qq[0] * (1.f / 65536.f) - mean * mean;
    float sc   = gamma[cI] * rsqrtf(var + 1e-5f);
    scaleA[cI] = sc;
    shiftA[cI] = beta[cI] - mean * sc;
  }
}

// 1x1 conv heads (36 reg + 9 cls channels) + sigmoid, batch 0 only
__global__ __launch_bounds__(256) void heads_kernel(
    const float* __restrict__ h0,
    const float* __restrict__ regw, const float* __restrict__ regb,
    const float* __restrict__ clsw, const float* __restrict__ clsb,
    float* __restrict__ regS, float* __restrict__ clsS)
{
  __shared__ float wsh[256];
  int ch = blockIdx.x;                      // 0..35 reg, 36..44 cls
  int p  = blockIdx.y * 256 + threadIdx.x;
  const float* wp; float bv; float* outp;
  if (ch < 36) { wp = regw + ch * 256;        bv = regb[ch];      outp = regS + (size_t)ch * 4096; }
  else         { int cc = ch - 36; wp = clsw + cc * 256; bv = clsb[cc]; outp = clsS + (size_t)cc * 4096; }
  wsh[threadIdx.x] = wp[threadIdx.x];
  __syncthreads();
  float acc = bv;
#pragma unroll 8
  for (int cI = 0; cI < 256; ++cI)
    acc += wsh[cI] * h0[((size_t)cI << 12) + p];
  outp[p] = 1.f / (1.f + expf(-acc));
}

// IoU of every anchor vs every GT box
__global__ __launch_bounds__(256) void iou_kernel(
    const float* __restrict__ gt, float* __restrict__ iouA)
{
  __shared__ float g[64];
  if (threadIdx.x < 64) g[threadIdx.x] = gt[threadIdx.x];
  __syncthreads();
  int n = blockIdx.x * 256 + threadIdx.x;   // 144*256 == 36864 exactly
  int cell = n / 9, a = n - cell * 9;
  float cy = (cell >> 6) + 0.5f, cx = (cell & 63) + 0.5f;
  float h = AH[a], w = AW[a];
  float y1 = cy - h * 0.5f, x1 = cx - w * 0.5f;
  float y2 = cy + h * 0.5f, x2 = cx + w * 0.5f;
  float aA = (y2 - y1) * (x2 - x1);
  for (int gi = 0; gi < NG; ++gi) {
    float gy1 = g[gi * 4], gx1 = g[gi * 4 + 1], gy2 = g[gi * 4 + 2], gx2 = g[gi * 4 + 3];
    float iy = fminf(y2, gy2) - fmaxf(y1, gy1);
    float ix = fminf(x2, gx2) - fmaxf(x1, gx1);
    float inter = fmaxf(iy, 0.f) * fmaxf(ix, 0.f);
    float aG = (gy2 - gy1) * (gx2 - gx1);
    iouA[n * NG + gi] = inter / (aA + aG - inter);
  }
}

// Per-GT argmax over anchors (first occurrence on ties)
__global__ __launch_bounds__(256) void argmax_kernel(
    const float* __restrict__ iouA, int* __restrict__ idxmax)
{
  __shared__ float bv[256];
  __shared__ int   bix[256];
  int g = blockIdx.x;
  float best = -1.f; int besti = 0x7fffffff;
  for (int n = threadIdx.x; n < NANC; n += 256) {
    float v = iouA[n * NG + g];
    if (v > best || (v == best && n < besti)) { best = v; besti = n; }
  }
  bv[threadIdx.x] = best; bix[threadIdx.x] = besti;
  __syncthreads();
  for (int off = 128; off > 0; off >>= 1) {
    if (threadIdx.x < off) {
      float ov = bv[threadIdx.x + off]; int oi = bix[threadIdx.x + off];
      if (ov > bv[threadIdx.x] || (ov == bv[threadIdx.x] && oi < bix[threadIdx.x])) {
        bv[threadIdx.x] = ov; bix[threadIdx.x] = oi;
      }
    }
    __syncthreads();
  }
  if (threadIdx.x == 0) idxmax[g] = bix[0];
}

__device__ __forceinline__ int neg_key(const float* __restrict__ iouA,
                                       const int* __restrict__ idxmax, int j)
{
  int g = j / NANC;
  int n = j - g * NANC;
  bool pos = (iouA[n * NG + g] > 0.5f) || (idxmax[g] == n);
  return pos ? 0 : 1;
}

// Per-block zero(=positive) counts for stable binary argsort
__global__ __launch_bounds__(256) void count_kernel(
    const float* __restrict__ iouA, const int* __restrict__ idxmax,
    int* __restrict__ bCnt)
{
  __shared__ int sh[256];
  int j0 = blockIdx.x * 1024 + threadIdx.x * 4;
  int z = 0;
#pragma unroll
  for (int i = 0; i < 4; ++i) z += (neg_key(iouA, idxmax, j0 + i) == 0);
  sh[threadIdx.x] = z;
  __syncthreads();
  for (int off = 128; off > 0; off >>= 1) {
    if (threadIdx.x < off) sh[threadIdx.x] += sh[threadIdx.x + off];
    __syncthreads();
  }
  if (threadIdx.x == 0) bCnt[blockIdx.x] = sh[0];
}

// Exclusive scan of 576 block counts (single block)
__global__ __launch_bounds__(1024) void scan_kernel(
    const int* __restrict__ bCnt, int* __restrict__ bOff)
{
  __shared__ int sh[1024];
  int tid = threadIdx.x;
  int v = (tid < 576) ? bCnt[tid] : 0;
  sh[tid] = v;
  __syncthreads();
  for (int off = 1; off < 1024; off <<= 1) {
    int t = (tid >= off) ? sh[tid - off] : 0;
    __syncthreads();
    sh[tid] += t;
    __syncthreads();
  }
  if (tid < 576) bOff[tid] = sh[tid] - v;
  if (tid == 575) bOff[576] = sh[575];    // total zeros
}

// Compute ranks (stable sort of binary key) and scatter all 5 outputs
__global__ __launch_bounds__(256) void scatter_kernel(
    const float* __restrict__ iouA, const int* __restrict__ idxmax,
    const int* __restrict__ bOff,
    const float* __restrict__ regS, const float* __restrict__ clsS,
    const float* __restrict__ gt, float* __restrict__ out)
{
  __shared__ int sh[256];
  __shared__ float gsh[64];
  int tid = threadIdx.x;
  if (tid < 64) gsh[tid] = gt[tid];
  int j0 = blockIdx.x * 1024 + tid * 4;
  int keys[4]; int cnt = 0;
#pragma unroll
  for (int i = 0; i < 4; ++i) { keys[i] = neg_key(iouA, idxmax, j0 + i); cnt += (keys[i] == 0); }
  sh[tid] = cnt;
  __syncthreads();
  for (int off = 1; off < 256; off <<= 1) {
    int t = (tid >= off) ? sh[tid - off] : 0;
    __syncthreads();
    sh[tid] += t;
    __syncthreads();
  }
  int excl   = sh[tid] - cnt;
  int zbase  = bOff[blockIdx.x] + excl;
  int totalZ = bOff[576];
  int z = 0;
#pragma unroll
  for (int i = 0; i < 4; ++i) {
    int j = j0 + i;
    int g = j / NANC;
    int n = j - g * NANC;
    int key = keys[i];
    int zb = zbase + z;              // zeros strictly before j
    int rank, match; float label;
    if (key == 0) { rank = zb; z++; label = 1.f; match = g; }
    else          { rank = totalZ + (j - zb); label = 0.f; match = 0; }

    // out0: reg[0, anc_idx]  (reg flat dim ordering: ch-subgroup major)
    int pix = n & 4095, csub = n >> 12;
#pragma unroll
    for (int k = 0; k < 4; ++k)
      out[(size_t)rank * 4 + k] = regS[((size_t)(k * 9 + csub) << 12) + pix];
    // out1: cls[0, anc_idx]
    out[2359296 + rank] = clsS[n];
    // out2: anc_box (cy,cx,h,w)/28 reconstructed exactly as reference
    int cell = n / 9, a = n - cell * 9;
    float cy = (cell >> 6) + 0.5f, cx = (cell & 63) + 0.5f;
    float hh = AH[a], ww = AW[a];
    float y1 = cy - hh * 0.5f, x1 = cx - ww * 0.5f;
    float y2 = cy + hh * 0.5f, x2 = cx + ww * 0.5f;
    float* ob = out + 2949120 + (size_t)rank * 4;
    ob[0] = ((y1 + y2) * 0.5f) / 28.f;
    ob[1] = ((x1 + x2) * 0.5f) / 28.f;
    ob[2] = (y2 - y1) / 28.f;
    ob[3] = (x2 - x1) / 28.f;
    // out3: anc_label
    out[5308416 + rank] = label;
    // out4: gt_per_anc
    float* og = out + 5898240 + (size_t)rank * 4;
#pragma unroll
    for (int k = 0; k < 4; ++k) og[k] = gsh[match * 4 + k];
  }
}

extern "C" void kernel_launch(void* const* d_in, const int* in_sizes, int n_in,
                              void* d_out, int out_size, void* d_ws, size_t ws_size,
                              hipStream_t stream) {
  const float* x     = (const float*)d_in[0];
  const float* gt    = (const float*)d_in[1];
  const float* cw    = (const float*)d_in[2];
  const float* cb    = (const float*)d_in[3];
  const float* gamma = (const float*)d_in[4];
  const float* beta  = (const float*)d_in[5];
  const float* regw  = (const float*)d_in[6];
  const float* regb  = (const float*)d_in[7];
  const float* clsw  = (const float*)d_in[8];
  const float* clsb  = (const float*)d_in[9];
  float* out = (float*)d_out;

  float* ws     = (float*)d_ws;
  float* sumP   = ws;                 // 256*1024
  float* sqP    = ws + 262144;        // 256*1024
  float* scaleA = ws + 524288;        // 256
  float* shiftA = ws + 524544;        // 256
  float* h0     = ws + 524800;        // 256*4096
  float* regS   = ws + 1573376;       // 36*4096
  float* clsS   = ws + 1720832;       // 9*4096
  float* iouA   = ws + 1757696;       // 36864*16
  int*   idxmax = (int*)(ws + 2347520);  // 16
  int*   bCnt   = (int*)(ws + 2347536);  // 576
  int*   bOff   = (int*)(ws + 2348112);  // 577

  // conv (all batches) -> per-channel sum/sumsq partials
  conv_wmma_kernel<<<2048, 256, 0, stream>>>(x, cw, cb, scaleA, shiftA,
                                             sumP, sqP, h0, 0);
  // BN scale/shift
  stats_kernel<<<256, 256, 0, stream>>>(sumP, sqP, gamma, beta, scaleA, shiftA);
  // conv batch 0 + BN + ReLU -> h0
  conv_wmma_kernel<<<128, 256, 0, stream>>>(x, cw, cb, scaleA, shiftA,
                                            sumP, sqP, h0, 1);
  // 1x1 heads + sigmoid
  heads_kernel<<<dim3(45, 16), 256, 0, stream>>>(h0, regw, regb, clsw, clsb,
                                                 regS, clsS);
  // anchor matching
  iou_kernel<<<144, 256, 0, stream>>>(gt, iouA);
  argmax_kernel<<<16, 256, 0, stream>>>(iouA, idxmax);
  count_kernel<<<576, 256, 0, stream>>>(iouA, idxmax, bCnt);
  scan_kernel<<<1, 1024, 0, stream>>>(bCnt, bOff);
  scatter_kernel<<<576, 256, 0, stream>>>(iouA, idxmax, bOff, regS, clsS, gt, out);
}